// Model_11888469475934
// MI455X (gfx1250) — compile-verified
//
#include <hip/hip_runtime.h>

// Fused pad -> conv1d(k=3) -> +bias -> maxpool(4,4)+argmax -> channel
// log_softmax -> scatter(ls*val) -> channel-sum -> crop, for
// B=32, C_IN=64, L=16376, C_MID=128. Conv cast as fp32 GEMM
// (M=128, K=192, N=B*16384) on v_wmma_f32_16x16x4_f32.
//
// LDS holds the im2col tile TRANSPOSED: btT[n][K], K = c_in*3+tap,
// stride 196. With identity K order both A and B fragments are two
// contiguous floats, so each is a single b64 load with an immediate
// offset directly into the WMMA operand pair: no pairing moves, no
// division, no address VALU in the inner loop. Unroll capped at 8 to
// stay under 256 VGPRs (no s_set_vgpr_msb churn).

typedef __attribute__((ext_vector_type(2))) float v2f;
typedef __attribute__((ext_vector_type(8))) float v8f;

#define L_IN_   16376
#define BSTR    196   // im2col row stride (192 used + 4): b64 pattern covers all 64 banks
#define CSTR    132   // conv tile stride (128 used + 4), bank-safe

__global__ __launch_bounds__(256)
void fused_conv_pool_lsm(const float* __restrict__ x,
                         const float* __restrict__ Wt,   // [128][64][3] = [128][192]
                         const float* __restrict__ bias, // [128]
                         float* __restrict__ out)        // [32][16376]
{
    // Phase 1: smem = im2col btT[n][K] (64 x 196 = 50176 B).
    // Phase 2: smem reused as conv[n][c_mid] (64 x 132 fits inside).
    __shared__ __align__(16) float smem[64 * BSTR];

    const int tile = blockIdx.x & 255;   // 256 tiles of 64 positions = 16384
    const int b    = blockIdx.x >> 8;
    const int g0   = tile << 6;
    const int tid  = threadIdx.x;

    // ---- stage im2col: btT[n][c*3+tap] = xp[c][n+tap] ----
    // xp[j] = x[j-3] for 3 <= j <= 16378, else 0 (covers the (3,5) pre-pad
    // and the conv's own (1,1) zero pad). Each loaded value fans out to
    // up to 3 LDS cells (predicated on 0 <= n < 64).
    for (int idx = tid; idx < 64 * 66; idx += 256) {
        int c  = idx / 66;
        int jj = idx - c * 66;           // padded coord j = g0 - 1 + jj
        int j  = g0 - 1 + jj;
        float v = 0.0f;
        if (j >= 3 && j <= 16378) v = x[(b * 64 + c) * L_IN_ + (j - 3)];
#pragma unroll
        for (int tap = 0; tap < 3; ++tap) {
            int n = jj - tap;
            if (n >= 0 && n < 64) smem[n * BSTR + c * 3 + tap] = v;
        }
    }
    __syncthreads();

    // ---- GEMM: D[c_mid][n] = sum_K W[c_mid][K] * btT[n][K] ----
    const int wave = tid >> 5;
    const int lane = tid & 31;
    const int l16  = lane & 15;
    const int half = lane >> 4;          // 0: lanes 0-15, 1: lanes 16-31
    const int mt0  = (wave >> 1) << 1;   // this wave: M tiles {mt0, mt0+1}
    const int nt0  = (wave & 1) << 1;    //             N tiles {nt0, nt0+1}

    v8f acc00 = {}, acc01 = {}, acc10 = {}, acc11 = {};
    const int n0 = nt0 * 16 + l16;
    const int n1 = n0 + 16;

    // Lane holds K = 4*kk + 2*half + {0,1}: contiguous pair for A and B.
    const float* A0 = Wt + (mt0 * 16 + l16) * 192 + 2 * half;
    const float* A1 = A0 + 16 * 192;
    const float* B0 = smem + n0 * BSTR + 2 * half;
    const float* B1 = smem + n1 * BSTR + 2 * half;

#pragma unroll 8
    for (int kk = 0; kk < 48; ++kk) {
        v2f a0 = *(const v2f*)(A0 + 4 * kk);
        v2f a1 = *(const v2f*)(A1 + 4 * kk);
        v2f b0 = *(const v2f*)(B0 + 4 * kk);
        v2f b1 = *(const v2f*)(B1 + 4 * kk);
        acc00 = __builtin_amdgcn_wmma_f32_16x16x4_f32(false, a0, false, b0, (short)0, acc00, false, false);
        acc01 = __builtin_amdgcn_wmma_f32_16x16x4_f32(false, a0, false, b1, (short)0, acc01, false, false);
        acc10 = __builtin_amdgcn_wmma_f32_16x16x4_f32(false, a1, false, b0, (short)0, acc10, false, false);
        acc11 = __builtin_amdgcn_wmma_f32_16x16x4_f32(false, a1, false, b1, (short)0, acc11, false, false);
    }

    // All waves must finish reading the im2col before it is overwritten.
    __syncthreads();

    // ---- write conv + bias to LDS (C/D layout: VGPR i -> M = mt*16 + half*8 + i, N = l16) ----
    {
        const int mb0 = mt0 * 16 + half * 8;
        const int mb1 = mb0 + 16;
        float* r00 = &smem[n0 * CSTR + mb0];
        float* r01 = &smem[n1 * CSTR + mb0];
        float* r10 = &smem[n0 * CSTR + mb1];
        float* r11 = &smem[n1 * CSTR + mb1];
#pragma unroll
        for (int i = 0; i < 8; ++i) {
            const float b0 = bias[mb0 + i];
            const float b1 = bias[mb1 + i];
            r00[i] = acc00[i] + b0;
            r01[i] = acc01[i] + b0;
            r10[i] = acc10[i] + b1;
            r11[i] = acc11[i] + b1;
        }
    }
    __syncthreads();

    // ---- pool(4) argmax + channel log_softmax + scatter(ls*val) + channel sum ----
    // 16 windows per tile; each wave owns 2 windows; lane owns channels lane+32j.
    for (int w = 0; w < 2; ++w) {
        const int p  = wave * 2 + w;
        const int nb = p * 4;
        float val[4]; int rel[4];
#pragma unroll
        for (int j = 0; j < 4; ++j) {
            const int c = lane + 32 * j;
            const float* cp = &smem[nb * CSTR + c];
            float best = cp[0]; int r = 0;
#pragma unroll
            for (int q = 1; q < 4; ++q) {           // strict > keeps FIRST max (jnp.argmax)
                float vq = cp[q * CSTR];
                if (vq > best) { best = vq; r = q; }
            }
            val[j] = best; rel[j] = r;
        }
        float m = fmaxf(fmaxf(val[0], val[1]), fmaxf(val[2], val[3]));
#pragma unroll
        for (int d = 16; d >= 1; d >>= 1) m = fmaxf(m, __shfl_xor(m, d, 32));
        float s = 0.0f;
#pragma unroll
        for (int j = 0; j < 4; ++j) s += __expf(val[j] - m);
#pragma unroll
        for (int d = 16; d >= 1; d >>= 1) s += __shfl_xor(s, d, 32);
        const float lg = __logf(s);

        float a0 = 0.f, a1 = 0.f, a2 = 0.f, a3 = 0.f;
#pragma unroll
        for (int j = 0; j < 4; ++j) {
            const float contrib = (val[j] - m - lg) * val[j];
            a0 += (rel[j] == 0) ? contrib : 0.0f;
            a1 += (rel[j] == 1) ? contrib : 0.0f;
            a2 += (rel[j] == 2) ? contrib : 0.0f;
            a3 += (rel[j] == 3) ? contrib : 0.0f;
        }
#pragma unroll
        for (int d = 16; d >= 1; d >>= 1) {
            a0 += __shfl_xor(a0, d, 32);
            a1 += __shfl_xor(a1, d, 32);
            a2 += __shfl_xor(a2, d, 32);
            a3 += __shfl_xor(a3, d, 32);
        }
        if (lane < 4) {
            const int g = g0 + nb + lane;           // padded coord; crop [3, 16378]
            if (g >= 3 && g <= 16378) {
                float r = (lane == 0) ? a0 : (lane == 1) ? a1 : (lane == 2) ? a2 : a3;
                out[b * L_IN_ + (g - 3)] = r;
            }
        }
    }
}

extern "C" void kernel_launch(void* const* d_in, const int* in_sizes, int n_in,
                              void* d_out, int out_size, void* d_ws, size_t ws_size,
                              hipStream_t stream) {
    (void)in_sizes; (void)n_in; (void)out_size; (void)d_ws; (void)ws_size;
    const float* x    = (const float*)d_in[0];   // [32][64][16376]
    const float* W    = (const float*)d_in[1];   // [128][64][3]
    const float* bias = (const float*)d_in[2];   // [128]
    float* out = (float*)d_out;                  // [32][1][16376]
    // 32 batches * 256 position-tiles
    fused_conv_pool_lsm<<<dim3(32 * 256), dim3(256), 0, stream>>>(x, W, bias, out);
}